// SVFTLayer_6493990552209
// MI455X (gfx1250) — compile-verified
//
#include <hip/hip_runtime.h>

// SVFT layer for MI455X (gfx1250), fp32 end-to-end via V_WMMA_F32_16X16X4_F32,
// with gfx1250 async direct-to-LDS staging (GLOBAL_LOAD_ASYNC_TO_LDS_B128 /
// ASYNCcnt) for the GEMM double buffer.
//
//   A  = M @ v            (M = 9-diagonal banded 4096x4096)  -> stored transposed: At[d][i]
//   W  = u @ A            (W[do][di] = sum_n u[do][n]*A[n][di])   NT-GEMM with WMMA
//   out= x @ W^T          (out[m][do] = sum_di x[m][di]*W[do][di]) NT-GEMM with WMMA
//
// Workspace: At (64MB) + W (64MB) = 128MB in d_ws.

typedef float v2f __attribute__((ext_vector_type(2)));
typedef float v8f __attribute__((ext_vector_type(8)));
typedef int   v4i_b128 __attribute__((__vector_size__(4 * sizeof(int))));

#define N_DIM 4096
#define DIN   4096
#define DOUT  4096
#define OFFD  4

#define AS1 __attribute__((address_space(1)))
#define AS3 __attribute__((address_space(3)))

#if defined(__gfx1250__) && __has_builtin(__builtin_amdgcn_global_load_async_to_lds_b128)
#define USE_ASYNC_LDS 1
#else
#define USE_ASYNC_LDS 0
#endif

#if USE_ASYNC_LDS
// 16B global -> LDS DMA, tracked by ASYNCcnt.
// Generic "__shared__" pointer low 32 bits == LDS byte offset (ISA: flat->LDS
// truncation), so a uintptr_t round-trip yields a valid AS(3) pointer.
__device__ __forceinline__ void async_load16(const float* gptr, float* lptr) {
  __builtin_amdgcn_global_load_async_to_lds_b128(
      (AS1 v4i_b128*)(uintptr_t)gptr, (AS3 v4i_b128*)(uintptr_t)lptr, 0, 0);
}
__device__ __forceinline__ void wait_async_le4() {
#if __has_builtin(__builtin_amdgcn_s_wait_asynccnt)
  __builtin_amdgcn_s_wait_asynccnt(4);
#else
  asm volatile("s_wait_asynccnt 0x4" ::: "memory");
#endif
}
__device__ __forceinline__ void wait_async_0() {
#if __has_builtin(__builtin_amdgcn_s_wait_asynccnt)
  __builtin_amdgcn_s_wait_asynccnt(0);
#else
  asm volatile("s_wait_asynccnt 0x0" ::: "memory");
#endif
}
#else
__device__ __forceinline__ void wait_async_le4() {}
__device__ __forceinline__ void wait_async_0() {}
#endif

// ---------------------------------------------------------------------------
// Kernel 1: banded product + transpose.
// At[d][i] = sum_{off=-4..4} ( sigmoid(gate)*s[bandidx(i,off)] + (off==0)*s_pre[i] ) * v[i+off][d]
// 64x64 tile per block, transpose through LDS so global reads AND writes are coalesced.
// ---------------------------------------------------------------------------
__global__ __launch_bounds__(256)
void banded_vt_kernel(const float* __restrict__ v,
                      const float* __restrict__ s_pre,
                      const float* __restrict__ s,
                      const float* __restrict__ gate,
                      float* __restrict__ At) {
  __shared__ float tile[64][65];          // +1 pad: conflict-free transposed reads
  const int i0 = blockIdx.x * 64;
  const int d0 = blockIdx.y * 64;
  const int t  = threadIdx.x;
  const float sig = 1.0f / (1.0f + __expf(-gate[0]));
  const int dd = t & 63;                  // column within tile (contiguous in memory)
  const int r0 = t >> 6;                  // 0..3

  for (int p = 0; p < 16; ++p) {
    const int ii = r0 + p * 4;
    const int i  = i0 + ii;
    float acc = 0.0f;
    int base = 0;                         // running offset into concatenated diagonals
    #pragma unroll
    for (int off = -OFFD; off <= OFFD; ++off) {
      const int j = i + off;              // column of M / row of v
      if (j >= 0 && j < N_DIM) {
        const int sidx = base + (off < 0 ? j : i);
        float coef = sig * s[sidx];
        if (off == 0) coef += s_pre[i];
        acc += coef * v[(size_t)j * DIN + d0 + dd];
      }
      base += N_DIM - (off < 0 ? -off : off);
    }
    tile[ii][dd] = acc;
  }
  __syncthreads();
  const int il = t & 63;                  // write contiguous along i
  for (int p = 0; p < 16; ++p) {
    const int dr = r0 + p * 4;
    At[(size_t)(d0 + dr) * N_DIM + i0 + il] = tile[il][dr];
  }
}

// ---------------------------------------------------------------------------
// NT GEMM with fp32 WMMA:  C[m][n] = sum_k A[m][k] * B[n][k]
// Both operands row-major, K contiguous -> A-frag and B-frag are ds_load_b64
// matching the documented 16x4 fp32 A layout (lane%16 = row, lane/16 picks
// K-pair {0,1}/{2,3}) and the analogous 4x16 B layout.
//
// Block: 256 thr = 8 waves as 4(M) x 2(N). Block tile 128x128, BK=16,
// wave tile 32x64 = 2x4 WMMA tiles. Double-buffered LDS staged via async
// direct-to-LDS b128 DMAs (4 per thread per chunk), row stride 18 dwords
// (8B aligned b64 frag loads, bank-conflict free across 16 lanes).
// ---------------------------------------------------------------------------
#define BM  128
#define BN  128
#define BK  16
#define LDT 18

__global__ __launch_bounds__(256)
void gemm_nt_wmma(const float* __restrict__ A,   // [M][K]
                  const float* __restrict__ B,   // [Nn][K]
                  float* __restrict__ C,         // [M][Nn]
                  int M, int Nn, int K) {
  __shared__ float As[2][BM * LDT];
  __shared__ float Bs[2][BN * LDT];

  const int tid   = threadIdx.x;
  const int lane  = tid & 31;
  const int wid   = tid >> 5;
  const int wm    = wid & 3;              // wave row   (4 -> 128/32)
  const int wn    = wid >> 2;             // wave col   (2 -> 128/64)
  const int blockM = blockIdx.y * BM;
  const int blockN = blockIdx.x * BN;
  const int lrow  = lane & 15;
  const int lhalf = lane >> 4;

  v8f acc[2][4];
  #pragma unroll
  for (int i = 0; i < 2; ++i)
    #pragma unroll
    for (int j = 0; j < 4; ++j)
      acc[i][j] = {};

  // Stage one 128x16 A tile + 128x16 B tile: 4 x b128 per thread.
  auto stage = [&](int kc, int buf) {
    const int k0  = kc * BK;
    const int c4  = tid & 3;              // which float4 in the 16-float row
    const int row = tid >> 2;             // 0..63
    #pragma unroll
    for (int p = 0; p < 2; ++p) {
      const int r = row + p * 64;
      const float* ga = A + (size_t)(blockM + r) * K + k0 + c4 * 4;
      const float* gb = B + (size_t)(blockN + r) * K + k0 + c4 * 4;
      float* la = &As[buf][r * LDT + c4 * 4];
      float* lb = &Bs[buf][r * LDT + c4 * 4];
#if USE_ASYNC_LDS
      async_load16(ga, la);
      async_load16(gb, lb);
#else
      const float4 va = *reinterpret_cast<const float4*>(ga);
      *reinterpret_cast<float2*>(la)     = make_float2(va.x, va.y);
      *reinterpret_cast<float2*>(la + 2) = make_float2(va.z, va.w);
      const float4 vb = *reinterpret_cast<const float4*>(gb);
      *reinterpret_cast<float2*>(lb)     = make_float2(vb.x, vb.y);
      *reinterpret_cast<float2*>(lb + 2) = make_float2(vb.z, vb.w);
#endif
    }
  };

  // 4 kk-steps of 8 WMMAs on the given buffer.
  auto compute = [&](int buf) {
    #pragma unroll
    for (int kk = 0; kk < BK / 4; ++kk) {
      v2f afrag[2], bfrag[4];
      #pragma unroll
      for (int i = 0; i < 2; ++i) {
        const float* p = &As[buf][(wm * 32 + i * 16 + lrow) * LDT + kk * 4 + 2 * lhalf];
        afrag[i] = *reinterpret_cast<const v2f*>(p);
      }
      #pragma unroll
      for (int j = 0; j < 4; ++j) {
        const float* p = &Bs[buf][(wn * 64 + j * 16 + lrow) * LDT + kk * 4 + 2 * lhalf];
        bfrag[j] = *reinterpret_cast<const v2f*>(p);
      }
      #pragma unroll
      for (int i = 0; i < 2; ++i)
        #pragma unroll
        for (int j = 0; j < 4; ++j)
          acc[i][j] = __builtin_amdgcn_wmma_f32_16x16x4_f32(
              false, afrag[i], false, bfrag[j],
              (short)0, acc[i][j], false, false);
    }
  };

  const int nk = K / BK;
  stage(0, 0);

  for (int kc = 0; kc < nk - 1; ++kc) {
    const int buf = kc & 1;
    stage(kc + 1, buf ^ 1);   // 4 new async ops behind the 4 for chunk kc
    wait_async_le4();         // in-order completion => chunk kc landed
    __syncthreads();          // all waves' chunk-kc DMAs landed
    compute(buf);
    __syncthreads();          // nobody still reads buf when next stage writes it
  }
  wait_async_0();             // last chunk's 4 DMAs
  __syncthreads();
  compute((nk - 1) & 1);

  // C/D layout: VGPR g -> row = g + 8*(lane>=16), col = lane%16
  #pragma unroll
  for (int i = 0; i < 2; ++i)
    #pragma unroll
    for (int j = 0; j < 4; ++j) {
      const int rbase = blockM + wm * 32 + i * 16 + 8 * lhalf;
      const int col   = blockN + wn * 64 + j * 16 + lrow;
      #pragma unroll
      for (int g = 0; g < 8; ++g)
        C[(size_t)(rbase + g) * Nn + col] = acc[i][j][g];
    }
}

// ---------------------------------------------------------------------------
extern "C" void kernel_launch(void* const* d_in, const int* in_sizes, int n_in,
                              void* d_out, int out_size, void* d_ws, size_t ws_size,
                              hipStream_t stream) {
  const float* x     = (const float*)d_in[0];   // [4,2048,4096]
  const float* u     = (const float*)d_in[1];   // [4096,4096]
  const float* v     = (const float*)d_in[2];   // [4096,4096]
  const float* s_pre = (const float*)d_in[3];   // [4096]
  const float* s     = (const float*)d_in[4];   // [36844]
  const float* gate  = (const float*)d_in[5];   // [1]
  // d_in[6]=row, d_in[7]=col unused (band indices derived analytically)

  float* At  = (float*)d_ws;                    // [DIN][N_DIM]   64MB
  float* W   = At + (size_t)DIN * N_DIM;        // [DOUT][DIN]    64MB
  float* out = (float*)d_out;                   // [8192][DOUT]

  dim3 blk(256);

  // 1) At = (M @ v)^T
  banded_vt_kernel<<<dim3(N_DIM / 64, DIN / 64), blk, 0, stream>>>(v, s_pre, s, gate, At);

  // 2) W[do][di] = sum_n u[do][n] * At[di][n]      (== weight^T)
  gemm_nt_wmma<<<dim3(DIN / BN, DOUT / BM), blk, 0, stream>>>(u, At, W, DOUT, DIN, N_DIM);

  // 3) out[m][do] = sum_di x[m][di] * W[do][di]
  const int Mrows = 4 * 2048;
  gemm_nt_wmma<<<dim3(DOUT / BN, Mrows / BM), blk, 0, stream>>>(x, W, out, Mrows, DOUT, DIN);
}